// LiteBoxNetLoss_67877663146547
// MI455X (gfx1250) — compile-verified
//
#include <hip/hip_runtime.h>
#include <hip/hip_bf16.h>
#include <math.h>

typedef __attribute__((ext_vector_type(2))) float v2f;
typedef __attribute__((ext_vector_type(4))) float v4f;
typedef __attribute__((ext_vector_type(8))) float v8f;

#define NACC 11
#define TPB 256
#define WPB (TPB / 32)

// Problem constants (B=32, C=10, H=192, W=320)
#define HW    61440          // 192*320
#define CSTR  (10 * HW)      // per-batch stride in floats
#define NPIX4 491520         // 32*HW/4

__device__ __forceinline__ float sl1(float d) {
    float ad = fabsf(d);
    return ad < 1.0f ? 0.5f * d * d : ad - 0.5f;
}

// ds_swizzle_b32, group-of-32 xor mode: lane l reads lane (l ^ xor_mask).
// offset = {1'b0, xor_mask[4:0], or_mask=0, and_mask=0x1f}
template <int IMM>
__device__ __forceinline__ float swz_xor_add(float v) {
    int t = __builtin_amdgcn_ds_swizzle(__float_as_int(v), IMM);
    return v + __int_as_float(t);
}

// Full-wave (32 lane) sum: one V_WMMA_F32_16X16X4_F32 does the 32->16 column
// fold (A = 16x4 all-ones; B = {x,0} puts lane n and lane n+16 in different
// K-rows of column n%16, so D[m][n] = x_n + x_{n+16} for every m), then four
// ds_swizzle xor folds finish the 16-column sum.  Requires EXEC all-ones.
// Result valid in every lane.
__device__ __forceinline__ float wave_sum_wmma(float x) {
    v2f a = {1.0f, 1.0f};
    v2f b = {x, 0.0f};
    v8f c = {};
    c = __builtin_amdgcn_wmma_f32_16x16x4_f32(false, a, false, b, (short)0, c,
                                              false, false);
    float y = c[0];                     // lane l holds x_{l%16} + x_{l%16+16}
    y = swz_xor_add<0x201F>(y);         // xor 8
    y = swz_xor_add<0x101F>(y);         // xor 4
    y = swz_xor_add<0x081F>(y);         // xor 2
    y = swz_xor_add<0x041F>(y);         // xor 1
    return y;
}

__global__ __launch_bounds__(TPB) void loss_partial_kernel(
    const float* __restrict__ re, const float* __restrict__ gt,
    float* __restrict__ ws, int nblocks)
{
    float acc[NACC];
#pragma unroll
    for (int j = 0; j < NACC; ++j) acc[j] = 0.0f;

    const int stride = gridDim.x * blockDim.x;
    for (int i = blockIdx.x * blockDim.x + threadIdx.x; i < NPIX4; i += stride) {
        const int p    = i << 2;
        const int bIdx = p / HW;
        const int hw   = p - bIdx * HW;
        const int base = bIdx * CSTR + hw;

        v4f rch[10], gch[10];
#pragma unroll
        for (int ch = 0; ch < 10; ++ch) {
            rch[ch] = *(const v4f*)(re + base + ch * HW);   // global_load_b128
            gch[ch] = *(const v4f*)(gt + base + ch * HW);
        }

#pragma unroll
        for (int u = 0; u < 4; ++u) {
            const float g0 = gch[0][u];
            const float r0 = rch[0][u];
            const float m0f  = (g0 >= 0.0f)  ? 1.0f : 0.0f;
            const float mvf  = (g0 == 1.0f)  ? 1.0f : 0.0f;
            const float posf = (g0 >= 0.1f)  ? 1.0f : 0.0f;     // implies m0
            const float negf = m0f * ((g0 < 0.1f) ? 1.0f : 0.0f);

            acc[0] += mvf;
            acc[1] += posf;

            const float safe = fminf(fmaxf(r0, 1e-6f), 1.0f - 1e-6f);
            const float dg   = g0 - r0;
            acc[2] += posf * dg * dg * logf(safe + 6e-8f);
            const float t  = 1.0f - g0;
            const float t2 = t * t;
            acc[3] += negf * r0 * r0 * logf(1.0f + 6e-8f - safe) * (t2 * t2);

            // mv-gated channel terms
            acc[4] += mvf * (sl1(rch[1][u] - gch[1][u]) + sl1(rch[2][u] - gch[2][u]));
            acc[5] += mvf * (sl1(rch[3][u] - gch[3][u]) + sl1(rch[6][u] - gch[6][u]));
            acc[6] += mvf * (sl1(rch[3][u] - gch[6][u]) + sl1(rch[6][u] - gch[3][u]));

            const float e44 = rch[4][u] - gch[4][u];
            const float e77 = rch[7][u] - gch[7][u];
            const float e55 = rch[5][u] - gch[5][u];
            const float e88 = rch[8][u] - gch[8][u];
            acc[7] += mvf * (e44 * e44 + e77 * e77 + e55 * e55 + e88 * e88);

            const float e47 = rch[4][u] - gch[7][u];
            const float e74 = rch[7][u] - gch[4][u];
            const float e58 = rch[5][u] - gch[8][u];
            const float e85 = rch[8][u] - gch[5][u];
            acc[8] += mvf * (e47 * e47 + e74 * e74 + e58 * e58 + e85 * e85);

            const float c1 = 1.0f - rch[5][u] * rch[5][u] - rch[4][u] * rch[4][u];
            const float c2 = 1.0f - rch[8][u] * rch[8][u] - rch[7][u] * rch[7][u];
            acc[9]  += mvf * (c1 * c1 + c2 * c2);
            acc[10] += mvf * sl1(rch[9][u] - gch[9][u]);
        }
    }

    // ---- wave-level reduction on the matrix path (EXEC is all-ones here) ----
    __shared__ float sacc[WPB][NACC];
    const int wave = threadIdx.x >> 5;
    const int lane = threadIdx.x & 31;

    float red[NACC];
#pragma unroll
    for (int j = 0; j < NACC; ++j) red[j] = wave_sum_wmma(acc[j]);

    if (lane == 0) {                       // single divergent region
#pragma unroll
        for (int j = 0; j < NACC; ++j) sacc[wave][j] = red[j];
    }
    __syncthreads();

    if (threadIdx.x < NACC) {
        float s = 0.0f;
#pragma unroll
        for (int w = 0; w < WPB; ++w) s += sacc[w][threadIdx.x];
        ws[threadIdx.x * nblocks + blockIdx.x] = s;
    }
}

__global__ __launch_bounds__(32) void loss_finalize_kernel(
    const float* __restrict__ ws, float* __restrict__ out, int nblocks)
{
    const int lane = threadIdx.x;
    float tot[NACC];

#pragma unroll
    for (int j = 0; j < NACC; ++j) {
        // Per-lane strided scalar sum (pipelined loads), then one WMMA fold.
        float s = 0.0f;
        for (int k = lane; k < nblocks; k += 32) s += ws[j * nblocks + k];
        tot[j] = wave_sum_wmma(s);
    }

    if (lane == 0) {
        const float nv       = tot[0];
        const float num_pos  = tot[1];
        const float pos_term = -tot[2];
        const float neg_term = -tot[3];
        const float focal = (num_pos == 0.0f) ? neg_term
                                              : (pos_term + neg_term) / num_pos;
        const float inv_nv  = 1.0f / nv;
        const float inv2_nv = 0.5f * inv_nv;

        const float confidence = 1.0f * focal;                 // CONF_W
        const float pos_loss   = 1.0f * tot[4] * inv2_nv;      // POS_W
        const float len1       = 0.1f * tot[5] * inv2_nv;      // LEN_W
        const float len2       = 0.1f * tot[6] * inv2_nv;
        const float trig1      = 1.0f * tot[7] * inv2_nv;      // TRIG_W
        const float trig2      = 1.0f * tot[8] * inv2_nv;
        const float const_loss = 0.5f * tot[9] * inv_nv;       // CONST_W
        const float height     = 0.1f * tot[10] * inv_nv;      // LEN_W

        const float dims = fminf(len1 + trig1, len2 + trig2) + height;
        out[0] = confidence + pos_loss + dims + const_loss;
    }
}

extern "C" void kernel_launch(void* const* d_in, const int* in_sizes, int n_in,
                              void* d_out, int out_size, void* d_ws, size_t ws_size,
                              hipStream_t stream)
{
    const float* re = (const float*)d_in[0];
    const float* gt = (const float*)d_in[1];
    float* out = (float*)d_out;
    float* ws  = (float*)d_ws;

    // Block count (multiple of 32 for the finalize stride) that fits scratch.
    // 1024 blocks x 11 floats = 45 KB.
    int nb = 1024;
    while (nb > 32 && (size_t)NACC * nb * sizeof(float) > ws_size) nb >>= 1;

    loss_partial_kernel<<<nb, TPB, 0, stream>>>(re, gt, ws, nb);
    loss_finalize_kernel<<<1, 32, 0, stream>>>(ws, out, nb);
}